// PCEN_14929306321327
// MI455X (gfx1250) — compile-verified
//
#include <hip/hip_runtime.h>
#include <stdint.h>

// PCEN: B=32, T=4000, C=128, fp32.
// out = (x / (1e-12 + ema)^alpha + delta)^(1/root) - delta^(1/root)
// ema_t = w*x_t + (1-w)*ema_{t-1},  ema_{-1} := x_0  (matches jax scan with init=x0)

constexpr int B    = 32;
constexpr int T    = 4000;
constexpr int C    = 128;
constexpr int SEG  = 32;          // time steps per segment
constexpr int NSEG = T / SEG;     // 125 (exact)
constexpr float FLOOR_EPS = 1e-12f;

// ---------------------------------------------------------------------------
// CDNA5 LDS addressing: generic pointers to LDS carry the workgroup-relative
// LDS byte address in the low 32 bits.
// ---------------------------------------------------------------------------
__device__ __forceinline__ uint32_t lds_off32(const void* p) {
  return (uint32_t)(uintptr_t)p;
}

// ---------------------------------------------------------------------------
// Async global<->LDS (ASYNCcnt) — fallback / portable path.
// INST_OFFSET is added to BOTH the LDS and global address, so a whole 16 KB
// tile copy needs one base address + 8 offset immediates (no per-round VALU).
// ---------------------------------------------------------------------------
__device__ __forceinline__ void tile_load_async(float* smem, const float* gbase, int tid) {
  const uint32_t ldsb = lds_off32(smem) + (uint32_t)tid * 16u;
  const unsigned long long g = (unsigned long long)(uintptr_t)(gbase + tid * 4);
  asm volatile(
      "global_load_async_to_lds_b128 %0, %1, off\n\t"
      "global_load_async_to_lds_b128 %0, %1, off offset:2048\n\t"
      "global_load_async_to_lds_b128 %0, %1, off offset:4096\n\t"
      "global_load_async_to_lds_b128 %0, %1, off offset:6144\n\t"
      "global_load_async_to_lds_b128 %0, %1, off offset:8192\n\t"
      "global_load_async_to_lds_b128 %0, %1, off offset:10240\n\t"
      "global_load_async_to_lds_b128 %0, %1, off offset:12288\n\t"
      "global_load_async_to_lds_b128 %0, %1, off offset:14336"
      :: "v"(ldsb), "v"(g) : "memory");
}

__device__ __forceinline__ void tile_store_async(float* gbase, const float* smem, int tid) {
  const uint32_t ldsb = lds_off32(smem) + (uint32_t)tid * 16u;
  const unsigned long long g = (unsigned long long)(uintptr_t)(gbase + tid * 4);
  asm volatile(
      "global_store_async_from_lds_b128 %1, %0, off\n\t"
      "global_store_async_from_lds_b128 %1, %0, off offset:2048\n\t"
      "global_store_async_from_lds_b128 %1, %0, off offset:4096\n\t"
      "global_store_async_from_lds_b128 %1, %0, off offset:6144\n\t"
      "global_store_async_from_lds_b128 %1, %0, off offset:8192\n\t"
      "global_store_async_from_lds_b128 %1, %0, off offset:10240\n\t"
      "global_store_async_from_lds_b128 %1, %0, off offset:12288\n\t"
      "global_store_async_from_lds_b128 %1, %0, off offset:14336"
      :: "v"(ldsb), "v"(g) : "memory");
}

__device__ __forceinline__ void wait_async_zero() {
  asm volatile("s_wait_asynccnt 0x0" ::: "memory");
}

// ---------------------------------------------------------------------------
// Tensor Data Mover path (TENSORcnt). Each wave DMAs one 8x128 f32 quarter of
// the 32x128 tile. D# packing per CDNA5 ISA Ch.8:
//   group0: [count=1], lds_addr, global_addr lo32, {addr[56:32], type=2<<30}
//   group1: data_size=2(4B); tensor_dim0=128; tensor_dim1=rows; tile_dim0=128;
//           tile_dim1=rows; tensor_dim0_stride=128
// ---------------------------------------------------------------------------
#if defined(__has_builtin)
#if __has_builtin(__builtin_amdgcn_tensor_load_to_lds) && \
    __has_builtin(__builtin_amdgcn_tensor_store_from_lds) && \
    __has_builtin(__builtin_amdgcn_s_wait_tensorcnt)
#define PCEN_USE_TDM 1
#endif
#endif

#ifdef PCEN_USE_TDM
typedef unsigned int u32x4 __attribute__((ext_vector_type(4)));
typedef int          i32x4 __attribute__((ext_vector_type(4)));
typedef int          i32x8 __attribute__((ext_vector_type(8)));

__device__ __forceinline__ void tdm_desc_2d(u32x4& g0, i32x8& g1,
                                            uint32_t lds_byte, const void* gptr,
                                            uint32_t rows, uint32_t cols) {
  const unsigned long long ga = (unsigned long long)(uintptr_t)gptr;
  g0[0] = 1u;                                   // count=1 (valid), user mode
  g0[1] = lds_byte;                             // lds_addr
  g0[2] = (uint32_t)ga;                         // global_addr[31:0]
  g0[3] = (uint32_t)((ga >> 32) & 0x01FFFFFFu)  // global_addr[56:32]
        | 0x80000000u;                          // type=2 ("image")
  g1[0] = 0x20000;                              // data_size=2 -> 4 bytes
  g1[1] = (int)(cols << 16);                    // tensor_dim0[15:0]
  g1[2] = (int)(rows << 16);                    // tensor_dim0[31:16]=0 | tensor_dim1[15:0]
  g1[3] = (int)(cols << 16);                    // tensor_dim1[31:16]=0 | tile_dim0
  g1[4] = (int)rows;                            // tile_dim1 | tile_dim2=0
  g1[5] = (int)cols;                            // tensor_dim0_stride lo32
  g1[6] = 0;                                    // stride0 hi16 | stride1 lo16
  g1[7] = 0;                                    // stride1 hi32
}

__device__ __forceinline__ void tdm_load_2d(uint32_t lds_byte, const void* gptr,
                                            uint32_t rows, uint32_t cols) {
  u32x4 g0; i32x8 g1;
  tdm_desc_2d(g0, g1, lds_byte, gptr, rows, cols);
  i32x4 z4 = {0, 0, 0, 0};
#if __clang_major__ >= 23
  i32x8 z8 = {0, 0, 0, 0, 0, 0, 0, 0};
  __builtin_amdgcn_tensor_load_to_lds(g0, g1, z4, z4, z8, 0);
#else
  __builtin_amdgcn_tensor_load_to_lds(g0, g1, z4, z4, 0);
#endif
}

__device__ __forceinline__ void tdm_store_2d(uint32_t lds_byte, void* gptr,
                                             uint32_t rows, uint32_t cols) {
  u32x4 g0; i32x8 g1;
  tdm_desc_2d(g0, g1, lds_byte, gptr, rows, cols);
  i32x4 z4 = {0, 0, 0, 0};
#if __clang_major__ >= 23
  i32x8 z8 = {0, 0, 0, 0, 0, 0, 0, 0};
  __builtin_amdgcn_tensor_store_from_lds(g0, g1, z4, z4, z8, 0);
#else
  __builtin_amdgcn_tensor_store_from_lds(g0, g1, z4, z4, 0);
#endif
}
#endif // PCEN_USE_TDM

// Unified tile movers: TDM when available, async-VMEM otherwise.
__device__ __forceinline__ void tile_in(float* smem, const float* gbase, int tid) {
#ifdef PCEN_USE_TDM
  const int wv = __builtin_amdgcn_readfirstlane((int)(threadIdx.x >> 5));
  // wave wv owns time rows [8*wv, 8*wv+8): 8*128 f32 = 4 KB
  tdm_load_2d(lds_off32(smem) + (uint32_t)wv * 4096u, gbase + wv * 1024, 8u, 128u);
  __builtin_amdgcn_s_wait_tensorcnt(0);
#else
  tile_load_async(smem, gbase, tid);
  wait_async_zero();
#endif
  __syncthreads();
}

__device__ __forceinline__ void tile_out(float* gbase, const float* smem, int tid) {
  __syncthreads();   // smem columns are produced by other lanes
#ifdef PCEN_USE_TDM
  const int wv = __builtin_amdgcn_readfirstlane((int)(threadIdx.x >> 5));
  tdm_store_2d(lds_off32(smem) + (uint32_t)wv * 4096u, gbase + wv * 1024, 8u, 128u);
  __builtin_amdgcn_s_wait_tensorcnt(0);  // LDS must stay live until DMA drains
#else
  tile_store_async(gbase, (float*)smem, tid);
  wait_async_zero();
#endif
}

// ---------- Kernel 1: per-segment local EMA contribution (zero start) ----------
__global__ void __launch_bounds__(C)
pcen_seg_reduce(const float* __restrict__ x, const float* __restrict__ w_in,
                float* __restrict__ hbuf) {
  const int c = threadIdx.x;
  const int s = blockIdx.x;
  const int b = blockIdx.y;
  __shared__ __align__(16) float tile[SEG * C];   // 16 KB

  const float* gbase = x + ((size_t)b * T + (size_t)s * SEG) * C;
  tile_in(tile, gbase, c);

  const float w = fminf(fmaxf(w_in[c], 0.0f), 1.0f);
  const float a = 1.0f - w;

  float h = 0.0f;
#pragma unroll 8
  for (int t = 0; t < SEG; ++t)
    h = fmaf(a, h, w * tile[t * C + c]);

  hbuf[((size_t)b * NSEG + s) * C + c] = h;
}

// ---------- Kernel 2: cross-segment scan (in place: h -> segment start state) ----------
__global__ void __launch_bounds__(C)
pcen_seg_scan(const float* __restrict__ x, const float* __restrict__ w_in,
              float* __restrict__ buf) {
  const int c = threadIdx.x;
  const int b = blockIdx.x;

  const float w = fminf(fmaxf(w_in[c], 0.0f), 1.0f);
  const float a = 1.0f - w;
  float d = a * a;  d = d * d;  d = d * d;  d = d * d;  d = d * d;  // a^32

  float state = x[(size_t)b * T * C + c];   // ema_{-1} = x[b,0,c]
  for (int s = 0; s < NSEG; ++s) {
    const size_t idx = ((size_t)b * NSEG + s) * C + c;
    const float h = buf[idx];
    buf[idx] = state;                        // start state entering segment s
    state = fmaf(d, state, h);               // state after segment s
  }
}

// ---------- Kernel 3: recompute EMA inside segment + pointwise, DMA out ----------
__global__ void __launch_bounds__(C)
pcen_apply(const float* __restrict__ x, const float* __restrict__ alpha_in,
           const float* __restrict__ delta_in, const float* __restrict__ root_in,
           const float* __restrict__ w_in, const float* __restrict__ buf,
           float* __restrict__ out) {
  const int c = threadIdx.x;
  const int s = blockIdx.x;
  const int b = blockIdx.y;
  __shared__ __align__(16) float tin [SEG * C];   // 16 KB
  __shared__ __align__(16) float tout[SEG * C];   // 16 KB

  const float* gin = x + ((size_t)b * T + (size_t)s * SEG) * C;
  tile_in(tin, gin, c);

  const float w        = fminf(fmaxf(w_in[c], 0.0f), 1.0f);
  const float a        = 1.0f - w;
  const float alpha    = fminf(alpha_in[c], 1.0f);
  const float root     = fmaxf(root_in[c], 1.0f);
  const float inv_root = 1.0f / root;
  const float delta    = delta_in[c];
  const float dpow     = powf(delta, inv_root);

  float state = buf[((size_t)b * NSEG + s) * C + c];

#pragma unroll 4
  for (int t = 0; t < SEG; ++t) {
    const float xv = tin[t * C + c];
    state = fmaf(a, state, w * xv);                             // serial EMA chain
    const float p = exp2f(-alpha * log2f(FLOOR_EPS + state));   // (eps+ema)^-alpha
    const float o = exp2f(inv_root * log2f(fmaf(xv, p, delta))) - dpow;
    tout[t * C + c] = o;
  }

  float* gout = out + ((size_t)b * T + (size_t)s * SEG) * C;
  tile_out(gout, tout, c);
}

// ---------- Fallback: fully sequential per-chain (used only if ws too small) ----------
__global__ void __launch_bounds__(C)
pcen_fallback(const float* __restrict__ x, const float* __restrict__ alpha_in,
              const float* __restrict__ delta_in, const float* __restrict__ root_in,
              const float* __restrict__ w_in, float* __restrict__ out) {
  const int c = threadIdx.x;
  const int b = blockIdx.x;

  const float w        = fminf(fmaxf(w_in[c], 0.0f), 1.0f);
  const float a        = 1.0f - w;
  const float alpha    = fminf(alpha_in[c], 1.0f);
  const float root     = fmaxf(root_in[c], 1.0f);
  const float inv_root = 1.0f / root;
  const float delta    = delta_in[c];
  const float dpow     = powf(delta, inv_root);

  float state = x[(size_t)b * T * C + c];
  for (int t = 0; t < T; ++t) {
    const size_t idx = ((size_t)b * T + t) * C + c;
    const float xv = x[idx];
    state = fmaf(a, state, w * xv);
    const float p = exp2f(-alpha * log2f(FLOOR_EPS + state));
    out[idx] = exp2f(inv_root * log2f(fmaf(xv, p, delta))) - dpow;
  }
}

// ---------- Launch ----------
extern "C" void kernel_launch(void* const* d_in, const int* in_sizes, int n_in,
                              void* d_out, int out_size, void* d_ws, size_t ws_size,
                              hipStream_t stream) {
  const float* x     = (const float*)d_in[0];
  const float* alpha = (const float*)d_in[1];
  const float* delta = (const float*)d_in[2];
  const float* root  = (const float*)d_in[3];
  const float* ew    = (const float*)d_in[4];
  float* out = (float*)d_out;

  const size_t need = (size_t)B * NSEG * C * sizeof(float);   // 2 MB
  if (ws_size >= need) {
    float* buf = (float*)d_ws;
    dim3 grid(NSEG, B);
    pcen_seg_reduce<<<grid, C, 0, stream>>>(x, ew, buf);
    pcen_seg_scan  <<<B,    C, 0, stream>>>(x, ew, buf);
    pcen_apply     <<<grid, C, 0, stream>>>(x, alpha, delta, root, ew, buf, out);
  } else {
    pcen_fallback<<<B, C, 0, stream>>>(x, alpha, delta, root, ew, out);
  }
}